// Encoder_Block_66022237274310
// MI455X (gfx1250) — compile-verified
//
#include <hip/hip_runtime.h>

typedef float v2f __attribute__((ext_vector_type(2)));
typedef float v8f __attribute__((ext_vector_type(8)));

#define BB 16
#define CC 128
#define TT 1024
#define DKD 64
#define CT (CC * TT)
#define LOG10000 9.210340371976184f
#define PI_2 1.5707963267948966f

__device__ inline v8f wmma4(v2f a, v2f b, v8f c) {
  // V_WMMA_F32_16X16X4_F32: D = A(16x4 f32) * B(4x16 f32) + C(16x16 f32)
  return __builtin_amdgcn_wmma_f32_16x16x4_f32(false, a, false, b, (short)0, c,
                                               false, false);
}

// ---------------- elementwise / reduction kernels ----------------

__global__ void posenc_add(const float* __restrict__ x, float* __restrict__ out) {
  int idx = blockIdx.x * blockDim.x + threadIdx.x;  // BB*CT threads
  int ct = idx % CT;
  int c = ct / TT;
  int t = ct % TT;
  float freq = expf(-(float)(c & ~1) * (LOG10000 / (float)CC));
  float pe = sinf((float)t * freq + (float)(c & 1) * PI_2);
  out[idx] = x[idx] + pe;
}

__global__ void batch_stats(const float* __restrict__ src, float* __restrict__ stats) {
  __shared__ float ssum[256];
  __shared__ float ssq[256];
  int b = blockIdx.x;
  int tid = threadIdx.x;
  const float* p = src + b * CT;
  float s = 0.f, q = 0.f;
  for (int e = tid; e < CT; e += 256) {
    float v = p[e];
    s += v;
    q += v * v;
  }
  ssum[tid] = s;
  ssq[tid] = q;
  __syncthreads();
  for (int st = 128; st > 0; st >>= 1) {
    if (tid < st) {
      ssum[tid] += ssum[tid + st];
      ssq[tid] += ssq[tid + st];
    }
    __syncthreads();
  }
  if (tid == 0) {
    stats[b] = ssum[0];
    stats[16 + b] = ssq[0];
  }
}

__global__ void ln_apply(const float* __restrict__ src, const float* __restrict__ g,
                         const float* __restrict__ bb, const float* __restrict__ stats,
                         float* __restrict__ dst) {
  int idx = blockIdx.x * blockDim.x + threadIdx.x;
  int b = idx / CT;
  int ct = idx % CT;
  float inv_n = 1.0f / (float)CT;
  float mu = stats[b] * inv_n;
  float var = stats[16 + b] * inv_n - mu * mu;
  float r = rsqrtf(var + 1e-5f);
  dst[idx] = (src[idx] - mu) * r * g[ct] + bb[ct];
}

__global__ void dwconv(const float* __restrict__ src, const float* __restrict__ w,
                       const float* __restrict__ bias, float* __restrict__ dst) {
  int idx = blockIdx.x * blockDim.x + threadIdx.x;
  int ct = idx % CT;
  int c = ct / TT;
  int t = ct % TT;
  const float* row = src + (idx - t);  // base of [b][c][0]
  float s = bias[c];
#pragma unroll
  for (int j = 0; j < 7; ++j) {
    int tt = t + j - 3;
    if (tt >= 0 && tt < TT) s += row[tt] * w[c * 7 + j];
  }
  dst[idx] = s;
}

// ---------------- WMMA GEMM kernels ----------------
// One wave owns a 16(M) x 64(N) strip: 4 accumulators, A fragment loaded once
// per K-step and reused by 4 WMMAs.
// A: lane holds row m=lane&15, k-pair base 2*(lane>>4). B: lane holds col
// n=lane&15, same k-pair. D: row = vgpr + 8*(lane>>4), col = lane&15.

// res[b,m,t] = relu( sum_c W[m,c]*Y[b,c,t] + bias[m] ) + res[b,m,t]
__global__ __launch_bounds__(128) void pw_gemm(const float* __restrict__ Y,
                                               const float* __restrict__ W,
                                               const float* __restrict__ bias,
                                               float* __restrict__ res) {
  int l = threadIdx.x & 31;
  int tile = blockIdx.x * 4 + (threadIdx.x >> 5);  // 2048 strips
  int b = tile >> 7;
  int rem = tile & 127;
  int m0 = (rem >> 4) << 4;
  int n0 = (rem & 15) << 6;
  int lo = l & 15;
  int kh = (l >> 4) << 1;
  const float* yb = Y + b * CT;
  v8f acc0 = {}, acc1 = {}, acc2 = {}, acc3 = {};
  for (int kk = 0; kk < CC; kk += 4) {
    int k0 = kk + kh;
    v2f a;
    a.x = W[(m0 + lo) * CC + k0];
    a.y = W[(m0 + lo) * CC + k0 + 1];
    const float* p0 = yb + k0 * TT + n0 + lo;
    const float* p1 = p0 + TT;
    v2f b0, b1, b2, b3;
    b0.x = p0[0];  b0.y = p1[0];
    b1.x = p0[16]; b1.y = p1[16];
    b2.x = p0[32]; b2.y = p1[32];
    b3.x = p0[48]; b3.y = p1[48];
    acc0 = wmma4(a, b0, acc0);
    acc1 = wmma4(a, b1, acc1);
    acc2 = wmma4(a, b2, acc2);
    acc3 = wmma4(a, b3, acc3);
  }
  float* rb = res + b * CT;
#pragma unroll
  for (int v = 0; v < 8; ++v) {
    int m = m0 + v + ((l >> 4) << 3);
    float bsv = bias[m];
    float* r = rb + m * TT + n0 + lo;
    r[0]  = fmaxf(acc0[v] + bsv, 0.f) + r[0];
    r[16] = fmaxf(acc1[v] + bsv, 0.f) + r[16];
    r[32] = fmaxf(acc2[v] + bsv, 0.f) + r[32];
    r[48] = fmaxf(acc3[v] + bsv, 0.f) + r[48];
  }
}

// dst[b,t,0:64] = sum_c X[b,c,t] * W[c,0:64]
__global__ __launch_bounds__(128) void xtw_gemm(const float* __restrict__ X,
                                                const float* __restrict__ W,
                                                float* __restrict__ dst) {
  int l = threadIdx.x & 31;
  int tile = blockIdx.x * 4 + (threadIdx.x >> 5);  // 1024 strips
  int b = tile >> 6;
  int t0 = (tile & 63) << 4;
  int lo = l & 15;
  int kh = (l >> 4) << 1;
  const float* xb = X + b * CT;
  v8f acc0 = {}, acc1 = {}, acc2 = {}, acc3 = {};
  for (int kk = 0; kk < CC; kk += 4) {
    int k0 = kk + kh;
    v2f a;
    a.x = xb[k0 * TT + t0 + lo];
    a.y = xb[(k0 + 1) * TT + t0 + lo];
    const float* w0 = W + k0 * DKD + lo;
    const float* w1 = w0 + DKD;
    v2f b0, b1, b2, b3;
    b0.x = w0[0];  b0.y = w1[0];
    b1.x = w0[16]; b1.y = w1[16];
    b2.x = w0[32]; b2.y = w1[32];
    b3.x = w0[48]; b3.y = w1[48];
    acc0 = wmma4(a, b0, acc0);
    acc1 = wmma4(a, b1, acc1);
    acc2 = wmma4(a, b2, acc2);
    acc3 = wmma4(a, b3, acc3);
  }
  float* db = dst + b * TT * DKD;
#pragma unroll
  for (int v = 0; v < 8; ++v) {
    int t = t0 + v + ((l >> 4) << 3);
    float* d = db + t * DKD + lo;
    d[0]  = acc0[v];
    d[16] = acc1[v];
    d[32] = acc2[v];
    d[48] = acc3[v];
  }
}

// Single-head attention for one 16-query tile per wave.
// Scores strip (16 x 1024) in LDS; masked softmax; then P @ V.
__global__ __launch_bounds__(32) void attn_kernel(const float* __restrict__ Q,
                                                  const float* __restrict__ Km,
                                                  const float* __restrict__ Vm,
                                                  const float* __restrict__ mask,
                                                  float* __restrict__ head) {
  __shared__ float S[16 * TT];  // 64 KB
  int b = blockIdx.x >> 6;
  int q0 = (blockIdx.x & 63) << 4;
  int l = threadIdx.x;
  int lo = l & 15;
  int kh = (l >> 4) << 1;
  const float* qb = Q + b * TT * DKD;
  const float* kb = Km + b * TT * DKD;
  const float* vb = Vm + b * TT * DKD;
  const float* mb = mask + b * TT;

  // Preload the Q-tile A fragments (reused for all 64 key tiles).
  v2f afrag[DKD / 4];
#pragma unroll
  for (int kk = 0; kk < DKD; kk += 4) {
    int k0 = kk + kh;
    afrag[kk >> 2].x = qb[(q0 + lo) * DKD + k0];
    afrag[kk >> 2].y = qb[(q0 + lo) * DKD + k0 + 1];
  }

  // S = (Q Kᵀ) * 1/sqrt(64), masked
  for (int t = 0; t < TT / 16; ++t) {
    v8f acc = {};
#pragma unroll
    for (int kk = 0; kk < DKD; kk += 4) {
      int k0 = kk + kh;
      v2f bv;
      bv.x = kb[(t * 16 + lo) * DKD + k0];
      bv.y = kb[(t * 16 + lo) * DKD + k0 + 1];
      acc = wmma4(afrag[kk >> 2], bv, acc);
    }
#pragma unroll
    for (int v = 0; v < 8; ++v) {
      int row = v + ((l >> 4) << 3);
      int col = t * 16 + lo;
      float hm = mb[col];
      float sv = acc[v] * 0.125f;
      S[row * TT + col] = sv * hm + (1.f - hm) * -1e30f;
    }
  }
  __syncthreads();

  // row softmax; wave halves split each row, combined via shfl_xor(16)
  {
    int row = lo;
    int c0 = (l >> 4) * (TT / 2);
    float mx = -3.4e38f;
    for (int c = 0; c < TT / 2; ++c) mx = fmaxf(mx, S[row * TT + c0 + c]);
    mx = fmaxf(mx, __shfl_xor(mx, 16, 32));
    float sum = 0.f;
    for (int c = 0; c < TT / 2; ++c) sum += expf(S[row * TT + c0 + c] - mx);
    sum += __shfl_xor(sum, 16, 32);
    float inv = 1.0f / sum;
    float qm = mask[b * TT + q0 + row];
    for (int c = 0; c < TT / 2; ++c) {
      int ix = row * TT + c0 + c;
      S[ix] = expf(S[ix] - mx) * inv * qm;
    }
  }
  __syncthreads();

  // head = P @ V : one LDS A-fragment feeds 4 N-accumulators
  {
    v8f acc0 = {}, acc1 = {}, acc2 = {}, acc3 = {};
    for (int kk = 0; kk < TT; kk += 4) {
      int k0 = kk + kh;
      v2f a;
      a.x = S[lo * TT + k0];
      a.y = S[lo * TT + k0 + 1];
      const float* v0p = vb + k0 * DKD + lo;
      const float* v1p = v0p + DKD;
      v2f b0, b1, b2, b3;
      b0.x = v0p[0];  b0.y = v1p[0];
      b1.x = v0p[16]; b1.y = v1p[16];
      b2.x = v0p[32]; b2.y = v1p[32];
      b3.x = v0p[48]; b3.y = v1p[48];
      acc0 = wmma4(a, b0, acc0);
      acc1 = wmma4(a, b1, acc1);
      acc2 = wmma4(a, b2, acc2);
      acc3 = wmma4(a, b3, acc3);
    }
    float* hb = head + b * TT * DKD;
#pragma unroll
    for (int v = 0; v < 8; ++v) {
      int t = q0 + v + ((l >> 4) << 3);
      float* h = hb + t * DKD + lo;
      h[0]  = acc0[v];
      h[16] = acc1[v];
      h[32] = acc2[v];
      h[48] = acc3[v];
    }
  }
}

// res[b,m,t] += sum_d (Wo[d,m]+Wo[d+64,m]) * head[b,t,d]
__global__ __launch_bounds__(128) void wo_gemm(const float* __restrict__ H,
                                               const float* __restrict__ Wo,
                                               float* __restrict__ res) {
  int l = threadIdx.x & 31;
  int tile = blockIdx.x * 4 + (threadIdx.x >> 5);  // 2048 strips
  int b = tile >> 7;
  int rem = tile & 127;
  int m0 = (rem >> 4) << 4;
  int n0 = (rem & 15) << 6;
  int lo = l & 15;
  int kh = (l >> 4) << 1;
  const float* hb = H + b * TT * DKD;
  v8f acc0 = {}, acc1 = {}, acc2 = {}, acc3 = {};
  for (int kk = 0; kk < DKD; kk += 4) {
    int k0 = kk + kh;
    v2f a;
    a.x = Wo[k0 * CC + m0 + lo] + Wo[(k0 + 64) * CC + m0 + lo];
    a.y = Wo[(k0 + 1) * CC + m0 + lo] + Wo[(k0 + 65) * CC + m0 + lo];
    const float* h0 = hb + (n0 + lo) * DKD + k0;
    v2f b0, b1, b2, b3;
    b0.x = h0[0];            b0.y = h0[1];
    b1.x = h0[16 * DKD];     b1.y = h0[16 * DKD + 1];
    b2.x = h0[32 * DKD];     b2.y = h0[32 * DKD + 1];
    b3.x = h0[48 * DKD];     b3.y = h0[48 * DKD + 1];
    acc0 = wmma4(a, b0, acc0);
    acc1 = wmma4(a, b1, acc1);
    acc2 = wmma4(a, b2, acc2);
    acc3 = wmma4(a, b3, acc3);
  }
  float* rb = res + b * CT;
#pragma unroll
  for (int v = 0; v < 8; ++v) {
    int m = m0 + v + ((l >> 4) << 3);
    float* r = rb + m * TT + n0 + lo;
    r[0]  = acc0[v] + r[0];
    r[16] = acc1[v] + r[16];
    r[32] = acc2[v] + r[32];
    r[48] = acc3[v] + r[48];
  }
}

// out[b,m,t] = relu( sum_c fc_w[m,c]*X[b,c,t] + fc_b[m] ) + res[b,m,t]
__global__ __launch_bounds__(128) void fc_gemm(const float* __restrict__ X,
                                               const float* __restrict__ W,
                                               const float* __restrict__ bias,
                                               const float* __restrict__ res,
                                               float* __restrict__ out) {
  int l = threadIdx.x & 31;
  int tile = blockIdx.x * 4 + (threadIdx.x >> 5);  // 2048 strips
  int b = tile >> 7;
  int rem = tile & 127;
  int m0 = (rem >> 4) << 4;
  int n0 = (rem & 15) << 6;
  int lo = l & 15;
  int kh = (l >> 4) << 1;
  const float* xb = X + b * CT;
  v8f acc0 = {}, acc1 = {}, acc2 = {}, acc3 = {};
  for (int kk = 0; kk < CC; kk += 4) {
    int k0 = kk + kh;
    v2f a;
    a.x = W[(m0 + lo) * CC + k0];
    a.y = W[(m0 + lo) * CC + k0 + 1];
    const float* p0 = xb + k0 * TT + n0 + lo;
    const float* p1 = p0 + TT;
    v2f b0, b1, b2, b3;
    b0.x = p0[0];  b0.y = p1[0];
    b1.x = p0[16]; b1.y = p1[16];
    b2.x = p0[32]; b2.y = p1[32];
    b3.x = p0[48]; b3.y = p1[48];
    acc0 = wmma4(a, b0, acc0);
    acc1 = wmma4(a, b1, acc1);
    acc2 = wmma4(a, b2, acc2);
    acc3 = wmma4(a, b3, acc3);
  }
  const float* rb = res + b * CT;
  float* ob = out + b * CT;
#pragma unroll
  for (int v = 0; v < 8; ++v) {
    int m = m0 + v + ((l >> 4) << 3);
    float bsv = bias[m];
    const float* r = rb + m * TT + n0 + lo;
    float* o = ob + m * TT + n0 + lo;
    o[0]  = fmaxf(acc0[v] + bsv, 0.f) + r[0];
    o[16] = fmaxf(acc1[v] + bsv, 0.f) + r[16];
    o[32] = fmaxf(acc2[v] + bsv, 0.f) + r[32];
    o[48] = fmaxf(acc3[v] + bsv, 0.f) + r[48];
  }
}

// ---------------- host launch ----------------

extern "C" void kernel_launch(void* const* d_in, const int* in_sizes, int n_in,
                              void* d_out, int out_size, void* d_ws, size_t ws_size,
                              hipStream_t stream) {
  (void)in_sizes; (void)n_in; (void)out_size; (void)ws_size;
  const float* x    = (const float*)d_in[0];
  const float* mask = (const float*)d_in[1];
  const float* dw_w = (const float*)d_in[2];
  const float* dw_b = (const float*)d_in[3];
  const float* pw_w = (const float*)d_in[4];
  const float* pw_b = (const float*)d_in[5];
  const float* n0g  = (const float*)d_in[6];
  const float* n0b  = (const float*)d_in[7];
  const float* nsg  = (const float*)d_in[8];
  const float* nsb  = (const float*)d_in[9];
  const float* neg_ = (const float*)d_in[10];
  const float* neb  = (const float*)d_in[11];
  const float* Wq   = (const float*)d_in[12];
  const float* Wk   = (const float*)d_in[13];
  const float* Wv   = (const float*)d_in[14];
  const float* Wo   = (const float*)d_in[15];
  const float* fcw  = (const float*)d_in[16];
  const float* fcb  = (const float*)d_in[17];
  float* outp = (float*)d_out;
  float* ws = (float*)d_ws;

  float* OUT = ws;                       // B*C*T
  float* RES = ws + 2097152;             // B*C*T
  float* TMP = ws + 4194304;             // B*C*T
  float* Qb  = ws + 6291456;             // B*T*64
  float* Kb  = ws + 7340032;
  float* Vb  = ws + 8388608;
  float* Hb  = ws + 9437184;
  float* ST  = ws + 10485760;            // 32 floats

  posenc_add<<<8192, 256, 0, stream>>>(x, OUT);
  batch_stats<<<16, 256, 0, stream>>>(OUT, ST);
  ln_apply<<<8192, 256, 0, stream>>>(OUT, n0g, n0b, ST, RES);

  for (int i = 0; i < 4; ++i) {
    dwconv<<<8192, 256, 0, stream>>>(OUT, dw_w + i * CC * 7, dw_b + i * CC, TMP);
    pw_gemm<<<512, 128, 0, stream>>>(TMP, pw_w + i * CC * CC, pw_b + i * CC, RES);
    batch_stats<<<16, 256, 0, stream>>>(RES, ST);
    ln_apply<<<8192, 256, 0, stream>>>(RES, nsg + i * CT, nsb + i * CT, ST, OUT);
  }

  xtw_gemm<<<256, 128, 0, stream>>>(OUT, Wq, Qb);
  xtw_gemm<<<256, 128, 0, stream>>>(OUT, Wk, Kb);
  xtw_gemm<<<256, 128, 0, stream>>>(OUT, Wv, Vb);
  attn_kernel<<<1024, 32, 0, stream>>>(Qb, Kb, Vb, mask, Hb);
  wo_gemm<<<512, 128, 0, stream>>>(Hb, Wo, RES);
  batch_stats<<<16, 256, 0, stream>>>(RES, ST);
  ln_apply<<<8192, 256, 0, stream>>>(RES, neg_, neb, ST, OUT);
  fc_gemm<<<512, 128, 0, stream>>>(OUT, fcw, fcb, RES, outp);
}